// RNN_onelayer_simplified_19378892439812
// MI455X (gfx1250) — compile-verified
//
#include <hip/hip_runtime.h>
#include <hip/hip_bf16.h>
#include <math.h>

// Problem dims (fixed by reference)
#define B_ 128
#define S_ 25
#define E_ 512
#define F_ 512
#define H_ 512
#define V_ 10000

typedef __attribute__((ext_vector_type(2))) float v2f;
typedef __attribute__((ext_vector_type(8))) float v8f;

// D = A(16x4) * B(4x16) + C, fp32, wave32.
__device__ __forceinline__ v8f wmma4(v2f a, v2f b, v8f c) {
  return __builtin_amdgcn_wmma_f32_16x16x4_f32(
      /*neg_a=*/false, a, /*neg_b=*/false, b,
      /*c_mod=*/(short)0, c, /*reuse_a=*/false, /*reuse_b=*/false);
}

// ---------------------------------------------------------------------------
// Kernel 1: U[b][h] = baseimgfeat[b,:] @ W_cell[0:F,:] + b_cell[h]
// M=128, N=512, K=512 -> 8x32 = 256 wave tiles, 8 waves/block -> 32 blocks
// ---------------------------------------------------------------------------
__global__ __launch_bounds__(256) void k_base(const float* __restrict__ A,
                                              const float* __restrict__ Wc,
                                              const float* __restrict__ bcell,
                                              float* __restrict__ U) {
  const int lane = threadIdx.x & 31;
  const int wave = threadIdx.x >> 5;
  const int tile = blockIdx.x * 8 + wave;
  const int TN = H_ / 16;
  const int tm = tile / TN, tn = tile % TN;
  const int row0 = tm * 16, col0 = tn * 16;
  const int half = (lane >> 4) & 1;
  const int l15 = lane & 15;

  v8f acc;
  const float cinit = bcell[col0 + l15];
#pragma unroll
  for (int i = 0; i < 8; ++i) acc[i] = cinit;

  const float* __restrict__ arow = A + (long)(row0 + l15) * F_;
#pragma unroll 8
  for (int k0 = 0; k0 < F_; k0 += 4) {
    const int kk = k0 + 2 * half;
    v2f a;
    a.x = arow[kk];
    a.y = arow[kk + 1];
    v2f b;
    b.x = Wc[(long)kk * H_ + col0 + l15];
    b.y = Wc[(long)(kk + 1) * H_ + col0 + l15];
    acc = wmma4(a, b, acc);
  }
#pragma unroll
  for (int r = 0; r < 8; ++r) {
    const int m = r + 8 * half;
    U[(long)(row0 + m) * H_ + col0 + l15] = acc[r];
  }
}

// ---------------------------------------------------------------------------
// Kernel 2: PRE[s*B+b][h] = U[b][h] + embedding[xTok[b][s],:] @ W_cell[F:F+E,:]
// M=3200, N=512, K=512. Register-blocked 2(M)x4(N) tiles per wave:
//   100 M-groups x 8 N-groups = 800 wave jobs -> 100 blocks x 8 waves.
// Gather of the embedding rows is fused into the A-fragment loads.
// ---------------------------------------------------------------------------
__global__ __launch_bounds__(256) void k_embed(const int* __restrict__ xTok,
                                               const float* __restrict__ Emb,
                                               const float* __restrict__ Wc,
                                               const float* __restrict__ U,
                                               float* __restrict__ PRE) {
  const int lane = threadIdx.x & 31;
  const int wave = threadIdx.x >> 5;
  const int job = blockIdx.x * 8 + wave;  // 800 jobs
  const int NG = (H_ / 16) / 4;           // 8 N-groups (64 cols each)
  const int mg = job / NG, ng = job % NG;
  const int row0 = mg * 32, col0 = ng * 64;
  const int half = (lane >> 4) & 1;
  const int l15 = lane & 15;

  // Gather token rows for the two M tiles (rows row0+l15 and row0+16+l15).
  const float* arow[2];
#pragma unroll
  for (int mi = 0; mi < 2; ++mi) {
    const int row = row0 + 16 * mi + l15;  // r = s*B + b
    const int s = row / B_;
    const int b = row & (B_ - 1);
    arow[mi] = Emb + (long)xTok[b * S_ + s] * E_;
  }

  v8f acc[2][4];
#pragma unroll
  for (int mi = 0; mi < 2; ++mi)
#pragma unroll
    for (int nj = 0; nj < 4; ++nj)
#pragma unroll
      for (int r = 0; r < 8; ++r) {
        const int b = (row0 + 16 * mi + r + 8 * half) & (B_ - 1);
        acc[mi][nj][r] = U[(long)b * H_ + col0 + 16 * nj + l15];
      }

  const float* __restrict__ Wce = Wc + (long)F_ * H_;  // rows F..F+E-1
#pragma unroll 2
  for (int k0 = 0; k0 < E_; k0 += 4) {
    const int kk = k0 + 2 * half;
    v2f a[2];
#pragma unroll
    for (int mi = 0; mi < 2; ++mi) {
      a[mi].x = arow[mi][kk];
      a[mi].y = arow[mi][kk + 1];
    }
#pragma unroll
    for (int nj = 0; nj < 4; ++nj) {
      v2f b;
      b.x = Wce[(long)kk * H_ + col0 + 16 * nj + l15];
      b.y = Wce[(long)(kk + 1) * H_ + col0 + 16 * nj + l15];
      acc[0][nj] = wmma4(a[0], b, acc[0][nj]);
      acc[1][nj] = wmma4(a[1], b, acc[1][nj]);
    }
  }

#pragma unroll
  for (int mi = 0; mi < 2; ++mi)
#pragma unroll
    for (int nj = 0; nj < 4; ++nj)
#pragma unroll
      for (int r = 0; r < 8; ++r) {
        const int m = 16 * mi + r + 8 * half;
        PRE[(long)(row0 + m) * H_ + col0 + 16 * nj + l15] = acc[mi][nj][r];
      }
}

// ---------------------------------------------------------------------------
// Kernel 3 (sequential, launched S times):
// Snew = tanh(PREs + Sprev @ W_cell[F+E : F+E+H , :])
// M=128, N=512, K=512 -> 256 wave tiles -> 32 blocks
// ---------------------------------------------------------------------------
__global__ __launch_bounds__(256) void k_step(const float* __restrict__ Sprev,
                                              const float* __restrict__ PREs,
                                              const float* __restrict__ Wh,
                                              float* __restrict__ Snew) {
  const int lane = threadIdx.x & 31;
  const int wave = threadIdx.x >> 5;
  const int tile = blockIdx.x * 8 + wave;
  const int TN = H_ / 16;
  const int tm = tile / TN, tn = tile % TN;
  const int row0 = tm * 16, col0 = tn * 16;
  const int half = (lane >> 4) & 1;
  const int l15 = lane & 15;

  v8f acc;
#pragma unroll
  for (int r = 0; r < 8; ++r) {
    const int m = r + 8 * half;
    acc[r] = PREs[(long)(row0 + m) * H_ + col0 + l15];
  }

  const float* __restrict__ arow = Sprev + (long)(row0 + l15) * H_;
#pragma unroll 8
  for (int k0 = 0; k0 < H_; k0 += 4) {
    const int kk = k0 + 2 * half;
    v2f a;
    a.x = arow[kk];
    a.y = arow[kk + 1];
    v2f b;
    b.x = Wh[(long)kk * H_ + col0 + l15];
    b.y = Wh[(long)(kk + 1) * H_ + col0 + l15];
    acc = wmma4(a, b, acc);
  }
#pragma unroll
  for (int r = 0; r < 8; ++r) {
    const int m = r + 8 * half;
    Snew[(long)(row0 + m) * H_ + col0 + l15] = tanhf(acc[r]);
  }
}

// ---------------------------------------------------------------------------
// Kernel 4: logits[b][s][v] = states[s*B+b,:] @ W_out + b_out[v]
// M=3200, N=10000 (=625*16), K=512. Register-blocked 2(M)x5(N) tiles/wave:
//   100 M-groups x 125 N-groups = 12500 wave jobs -> 3125 blocks x 4 waves.
// Per K-step of 4: 10 WMMAs fed by 2 A-float2 + 10 B-b32 loads.
// ---------------------------------------------------------------------------
__global__ __launch_bounds__(128) void k_logits(const float* __restrict__ St,
                                                const float* __restrict__ Wo,
                                                const float* __restrict__ bo,
                                                float* __restrict__ out) {
  const int lane = threadIdx.x & 31;
  const int wave = threadIdx.x >> 5;
  const int job = blockIdx.x * 4 + wave;  // 12500 jobs
  const int NG = 125;                     // N-groups of 80 cols
  const int mg = job / NG, ng = job % NG;
  const int row0 = mg * 32, col0 = ng * 80;
  const int half = (lane >> 4) & 1;
  const int l15 = lane & 15;

  v8f acc[2][5];
#pragma unroll
  for (int nj = 0; nj < 5; ++nj) {
    const float cinit = bo[col0 + 16 * nj + l15];
#pragma unroll
    for (int r = 0; r < 8; ++r) {
      acc[0][nj][r] = cinit;
      acc[1][nj][r] = cinit;
    }
  }

  const float* __restrict__ arow0 = St + (long)(row0 + l15) * H_;
  const float* __restrict__ arow1 = St + (long)(row0 + 16 + l15) * H_;
#pragma unroll 2
  for (int k0 = 0; k0 < H_; k0 += 4) {
    const int kk = k0 + 2 * half;
    v2f a0, a1;
    a0.x = arow0[kk];
    a0.y = arow0[kk + 1];
    a1.x = arow1[kk];
    a1.y = arow1[kk + 1];
    const float* __restrict__ brow0 = Wo + (long)kk * V_ + col0 + l15;
    const float* __restrict__ brow1 = brow0 + V_;
#pragma unroll
    for (int nj = 0; nj < 5; ++nj) {
      v2f b;
      b.x = brow0[16 * nj];
      b.y = brow1[16 * nj];
      acc[0][nj] = wmma4(a0, b, acc[0][nj]);
      acc[1][nj] = wmma4(a1, b, acc[1][nj]);
    }
  }

#pragma unroll
  for (int mi = 0; mi < 2; ++mi)
#pragma unroll
    for (int r = 0; r < 8; ++r) {
      const int row = row0 + 16 * mi + r + 8 * half;  // = s*B + b
      const int s = row / B_;
      const int b = row & (B_ - 1);
      float* __restrict__ orow = out + (long)b * S_ * V_ + (long)s * V_;
#pragma unroll
      for (int nj = 0; nj < 5; ++nj)
        orow[col0 + 16 * nj + l15] = acc[mi][nj][r];
    }
}

extern "C" void kernel_launch(void* const* d_in, const int* in_sizes, int n_in,
                              void* d_out, int out_size, void* d_ws, size_t ws_size,
                              hipStream_t stream) {
  (void)in_sizes; (void)n_in; (void)out_size; (void)ws_size;

  const int*   xTok    = (const int*)d_in[0];    // [B,S]
  const float* baseimg = (const float*)d_in[1];  // [B,F]
  const float* h0      = (const float*)d_in[2];  // [1,B,H]
  const float* emb     = (const float*)d_in[3];  // [V,E]
  const float* Wc      = (const float*)d_in[4];  // [F+E+H, H]
  const float* bcell   = (const float*)d_in[5];  // [1,H]
  const float* Wo      = (const float*)d_in[6];  // [H,V]
  const float* bo      = (const float*)d_in[7];  // [V]
  float* out = (float*)d_out;                    // [B,S,V] ++ [1,B,H]

  // Workspace layout (fp32): U[B*H] | PRE[S*B*H] | ST[S*B*H]  (~13.4 MB)
  float* U   = (float*)d_ws;
  float* PRE = U + (size_t)B_ * H_;
  float* ST  = PRE + (size_t)S_ * B_ * H_;

  k_base<<<32, 256, 0, stream>>>(baseimg, Wc, bcell, U);
  k_embed<<<100, 256, 0, stream>>>(xTok, emb, Wc, U, PRE);

  const float* Wh = Wc + (size_t)(F_ + E_) * H_;
  for (int s = 0; s < S_; ++s) {
    const float* prev = (s == 0) ? h0 : ST + (size_t)(s - 1) * B_ * H_;
    k_step<<<32, 256, 0, stream>>>(prev, PRE + (size_t)s * B_ * H_, Wh,
                                   ST + (size_t)s * B_ * H_);
  }

  k_logits<<<3125, 128, 0, stream>>>(ST, Wo, bo, out);

  // final_state [1,B,H] appended after logits
  hipMemcpyAsync(out + (size_t)B_ * S_ * V_, ST + (size_t)(S_ - 1) * B_ * H_,
                 (size_t)B_ * H_ * sizeof(float), hipMemcpyDeviceToDevice,
                 stream);
}